// lossFunc_180388626880
// MI455X (gfx1250) — compile-verified
//
#include <hip/hip_runtime.h>
#include <stdint.h>

// Problem constants (from the reference): B=1024, T=200, Q=512.
constexpr int Bn  = 1024;
constexpr int Tt  = 200;
constexpr int Qh  = 512;          // questions (half of batch's last dim)
constexpr int RF  = 2 * Qh;       // 1024 floats per batch row (4 KB)

constexpr int WPB     = 4;        // waves per block (wave32)
constexpr int THREADS = WPB * 32; // 128
constexpr int NBLK    = 4096;     // 16384 waves total

// -----------------------------------------------------------------------------
// Kernel 1: each wave owns a CONTIGUOUS run of (b,t) rows. Batch rows are
// double-buffered into LDS with CDNA5 async copies (GLOBAL_LOAD_ASYNC_TO_LDS,
// ASYNCcnt-tracked, non-temporal since the stream has zero reuse), scanned for
// the one-hot, then pred is sparsely gathered and BCE accumulated.
// -----------------------------------------------------------------------------
__global__ __launch_bounds__(THREADS)
void dkt_scan_kernel(const float* __restrict__ pred,
                     const float* __restrict__ batch,
                     float* __restrict__ partial,
                     int numRows, int rowsPerWave)
{
    // WPB waves * 2 buffers * 4KB each, plus WPB reduction slots at the tail.
    __shared__ __align__(16) float lds[WPB * 2 * RF + WPB];

    const int lane  = threadIdx.x & 31;
    const int wslot = threadIdx.x >> 5;
    const int wid   = blockIdx.x * WPB + wslot;

    float acc = 0.0f;

    // Issue one 4KB batch row (8 x 16B per lane) into this wave's LDS buffer.
    auto issue_row = [&](int b, int t1, int bufsel) {
        const float* src = batch + ((size_t)b * Tt + t1) * RF;
        unsigned ldsBase =
            (unsigned)(uintptr_t)(&lds[(wslot * 2 + bufsel) * RF]);
#pragma unroll
        for (int c = 0; c < 8; ++c) {
            int fidx = (c * 32 + lane) * 4;              // float index in row
            unsigned long long g = (unsigned long long)(uintptr_t)(src + fidx);
            unsigned l = ldsBase + (unsigned)fidx * 4u;  // LDS byte offset
            asm volatile(
                "global_load_async_to_lds_b128 %0, %1, off th:TH_LOAD_NT"
                :: "v"(l), "v"(g) : "memory");
        }
    };

    const int rowBeg = wid * rowsPerWave;
    const int rowEnd = min(rowBeg + rowsPerWave, numRows);

    if (rowBeg < rowEnd) {
        // (b, t1) of the first owned row; advanced incrementally afterwards.
        int cb = rowBeg / (Tt - 1);
        int ct = rowBeg - cb * (Tt - 1) + 1;             // t1 in [1, Tt)

        int buf = 0;
        issue_row(cb, ct, 0);

        for (int row = rowBeg; row < rowEnd; ++row) {
            // Next row's coordinates (contiguous stream, mostly same b).
            int nb = cb, nt = ct + 1;
            if (nt == Tt) { nt = 1; ++nb; }

            if (row + 1 < rowEnd) {
                issue_row(nb, nt, buf ^ 1);              // prefetch next 4KB
                // Drain the oldest 8 async ops (current buffer); keep 8 live.
                asm volatile("s_wait_asynccnt 8" ::: "memory");
            } else {
                asm volatile("s_wait_asynccnt 0" ::: "memory");
            }

            // Scan the 4KB row in LDS: locate the single nonzero element.
            const float4* lrow =
                (const float4*)(&lds[(wslot * 2 + buf) * RF]);
            int found = -1;
#pragma unroll
            for (int c = 0; c < 8; ++c) {
                int ci = c * 32 + lane;
                float4 v = lrow[ci];                     // ds_load_b128
                int j = ci * 4;
                if (v.x != 0.0f) found = j;
                if (v.y != 0.0f) found = j + 1;
                if (v.z != 0.0f) found = j + 2;
                if (v.w != 0.0f) found = j + 3;
            }

            // packed: bit30 = valid, bit16 = correct(first half), [8:0] = qid
            int packed = (found >= 0)
                ? ((1 << 30) | ((found < Qh) ? (1 << 16) : 0)
                   | (found & (Qh - 1)))
                : 0;
#pragma unroll
            for (int off = 16; off >= 1; off >>= 1)
                packed = max(packed, __shfl_xor(packed, off, 32));

            if (lane == 0 && (packed >> 30)) {
                int qid = packed & (Qh - 1);
                int a   = (packed >> 16) & 1;
                // sparse gather: p = pred[b, t1-1, qid]
                float p = pred[((size_t)cb * Tt + (ct - 1)) * Qh + qid];
                if (p > 0.0f)
                    acc += a ? logf(p) : logf(1.0f - p); // bce term
            }

            cb = nb; ct = nt;
            buf ^= 1;
        }
    }

    // Deterministic in-block reduction of the WPB wave accumulators.
    if (lane == 0) lds[WPB * 2 * RF + wslot] = acc;
    __syncthreads();
    if (threadIdx.x == 0) {
        float t = 0.0f;
        for (int i = 0; i < WPB; ++i) t += lds[WPB * 2 * RF + i];
        partial[blockIdx.x] = t;                         // every block writes
    }
}

// -----------------------------------------------------------------------------
// Kernel 2: deterministic fixed-order reduction of block partials -> loss.
// Double-precision accumulation for headroom; output stays fp32.
// -----------------------------------------------------------------------------
__global__ __launch_bounds__(256)
void dkt_reduce_kernel(const float* __restrict__ partial, int n,
                       float* __restrict__ out)
{
    __shared__ double s[256];
    double a = 0.0;
    for (int i = threadIdx.x; i < n; i += 256) a += (double)partial[i];
    s[threadIdx.x] = a;
    __syncthreads();
    if (threadIdx.x == 0) {
        double t = 0.0;
        for (int i = 0; i < 256; ++i) t += s[i];
        out[0] = (float)(-t);                            // loss = -sum(bce)
    }
}

// -----------------------------------------------------------------------------
extern "C" void kernel_launch(void* const* d_in, const int* in_sizes, int n_in,
                              void* d_out, int out_size, void* d_ws, size_t ws_size,
                              hipStream_t stream)
{
    (void)in_sizes; (void)n_in; (void)out_size;
    const float* pred  = (const float*)d_in[0];   // [B, T, Q] fp32
    const float* batch = (const float*)d_in[1];   // [B, T, 2Q] fp32
    float* partial = (float*)d_ws;                // NBLK floats (16 KB)
    float* out     = (float*)d_out;               // scalar loss

    int nblk = NBLK;
    if (ws_size < (size_t)nblk * sizeof(float))
        nblk = (int)(ws_size / sizeof(float));
    if (nblk < 1) nblk = 1;

    const int numRows    = Bn * (Tt - 1);         // 203,776 rows to scan
    const int totalWaves = nblk * WPB;
    const int rowsPerWave = (numRows + totalWaves - 1) / totalWaves;

    dkt_scan_kernel<<<dim3(nblk), dim3(THREADS), 0, stream>>>(
        pred, batch, partial, numRows, rowsPerWave);
    dkt_reduce_kernel<<<dim3(1), dim3(256), 0, stream>>>(partial, nblk, out);
}